// GATTP_1_14903536517939
// MI455X (gfx1250) — compile-verified
//
#include <hip/hip_runtime.h>

typedef __attribute__((ext_vector_type(2))) float v2f;
typedef __attribute__((ext_vector_type(8))) float v8f;

#define NSEG 1024
#define DIM  64
#define NH   32

// ---------------------------------------------------------------------------
// Kernel 1: find start offset of each segment in the sorted batch vector.
// seg_start[b] = lower_bound(batch, b), seg_start[NSEG] = N.  (1025 ints in ws)
// ---------------------------------------------------------------------------
__global__ void seg_bounds_kernel(const int* __restrict__ batch, int N,
                                  int* __restrict__ seg_start) {
    int b = blockIdx.x * blockDim.x + threadIdx.x;
    if (b > NSEG) return;
    if (b == NSEG) { seg_start[NSEG] = N; return; }
    int lo = 0, hi = N;
    while (lo < hi) {
        int mid = (lo + hi) >> 1;
        if (batch[mid] < b) lo = mid + 1; else hi = mid;
    }
    seg_start[b] = lo;
}

// float atomic-max on LDS via ordered-int trick (init value must be -inf)
__device__ __forceinline__ void atomic_max_f32_lds(float* addr, float val) {
    if (val >= 0.0f) atomicMax((int*)addr, __float_as_int(val));
    else             atomicMin((unsigned int*)addr, __float_as_uint(val));
}

// ---------------------------------------------------------------------------
// Kernel 2: one workgroup (8 wave32) per segment.
// Phase A: WMMA gates -> per-head segment max (LDS)
// Phase B: WMMA gates -> exp(g - max) -> per-head denom (LDS)
// Phase C: WMMA gates -> wn -> head-mean (shfl) -> pooled += wbar * x
// ---------------------------------------------------------------------------
__global__ __launch_bounds__(256) void gattp_main_kernel(
    const float* __restrict__ x, const float* __restrict__ Wg,
    const float* __restrict__ bg, const int* __restrict__ seg_start,
    float* __restrict__ out)
{
    __shared__ float sWg[DIM * NH];   // gate weights, [d][h] row-major (8 KB)
    __shared__ float sbg[NH];
    __shared__ float smax[NH];
    __shared__ float sden[NH];
    __shared__ float spool[DIM];
    __shared__ float swbar[8][16];    // per-wave row-weight staging

    const int seg  = blockIdx.x;
    const int tid  = threadIdx.x;
    const int lane = tid & 31;
    const int wave = tid >> 5;
    const int l16  = lane & 15;
    const int half = lane >> 4;       // 0 or 1 (lane group)

    const int start  = seg_start[seg];
    const int end    = seg_start[seg + 1];
    const int ntiles = (end - start + 15) >> 4;

    for (int i = tid; i < DIM * NH; i += 256) sWg[i] = Wg[i];
    if (tid < NH)  { sbg[tid] = bg[tid]; smax[tid] = -__builtin_inff(); sden[tid] = 0.f; }
    if (tid < DIM) spool[tid] = 0.f;
    __syncthreads();

    const float bias0 = sbg[l16];
    const float bias1 = sbg[16 + l16];

    // Compute biased gates for a 16-row tile into two 16x16 f32 accumulators:
    // c0 = heads 0..15, c1 = heads 16..31.  A-frag: row=l16, K=2*half+{0,1}.
    // Tail rows are CLAMPED to end-1 (valid memory); the resulting garbage
    // gates live only in C-rows >= end, which every consumer masks out.
    auto compute_gates = [&](int rowbase, bool pf, v8f& c0, v8f& c1) {
        v8f acc0 = {0.f,0.f,0.f,0.f,0.f,0.f,0.f,0.f};
        v8f acc1 = {0.f,0.f,0.f,0.f,0.f,0.f,0.f,0.f};
        int arow = rowbase + l16;
        if (arow >= end) arow = end - 1;                 // clamp, no exec masking
        const v2f* __restrict__ xr =
            (const v2f*)(x + (size_t)arow * DIM) + half; // 8B-aligned (row=256B)
        if (pf) {                                        // uniform guard: next tile
            int prow = arow + 8 * 16;
            if (prow >= end) prow = end - 1;
            __builtin_prefetch(x + (size_t)prow * DIM, 0, 3);
        }
        // Batch all 16 A-fragment loads first (distinct regs -> one load clause,
        // drained with partial s_wait_loadcnt while the WMMA chain retires).
        v2f a[16];
        #pragma unroll
        for (int kc = 0; kc < 16; ++kc) a[kc] = xr[kc * 2];
        #pragma unroll
        for (int kc = 0; kc < 16; ++kc) {
            const int k0 = kc * 4 + half * 2;            // B-frag rows (K)
            v2f b0v, b1v;
            b0v.x = sWg[(k0    ) * NH + l16];
            b0v.y = sWg[(k0 + 1) * NH + l16];
            b1v.x = sWg[(k0    ) * NH + 16 + l16];
            b1v.y = sWg[(k0 + 1) * NH + 16 + l16];
            acc0 = __builtin_amdgcn_wmma_f32_16x16x4_f32(false, a[kc], false, b0v,
                                                         (short)0, acc0, false, false);
            acc1 = __builtin_amdgcn_wmma_f32_16x16x4_f32(false, a[kc], false, b1v,
                                                         (short)0, acc1, false, false);
        }
        #pragma unroll
        for (int i = 0; i < 8; ++i) { acc0[i] += bias0; acc1[i] += bias1; }
        c0 = acc0; c1 = acc1;
    };

    // ---------------- Phase A: per-head segment max ----------------
    for (int t = wave; t < ntiles; t += 8) {
        const int rowbase = start + t * 16;
        v8f c0, c1;
        compute_gates(rowbase, t + 8 < ntiles, c0, c1);
        float m0 = -__builtin_inff(), m1 = -__builtin_inff();
        #pragma unroll
        for (int i = 0; i < 8; ++i) {                // row of VGPR i = i + 8*half
            if (rowbase + i + 8 * half < end) {
                m0 = fmaxf(m0, c0[i]);
                m1 = fmaxf(m1, c1[i]);
            }
        }
        if (m0 > -__builtin_inff()) atomic_max_f32_lds(&smax[l16], m0);
        if (m1 > -__builtin_inff()) atomic_max_f32_lds(&smax[16 + l16], m1);
    }
    __syncthreads();

    const float g0max = smax[l16];
    const float g1max = smax[16 + l16];

    // ---------------- Phase B: per-head denom (sum of exp) ----------------
    for (int t = wave; t < ntiles; t += 8) {
        const int rowbase = start + t * 16;
        v8f c0, c1;
        compute_gates(rowbase, t + 8 < ntiles, c0, c1);
        float s0 = 0.f, s1 = 0.f;
        #pragma unroll
        for (int i = 0; i < 8; ++i) {
            if (rowbase + i + 8 * half < end) {
                s0 += __expf(c0[i] - g0max);
                s1 += __expf(c1[i] - g1max);
            }
        }
        atomicAdd(&sden[l16], s0);
        atomicAdd(&sden[16 + l16], s1);
    }
    __syncthreads();

    const float rd0 = 1.f / sden[l16];
    const float rd1 = 1.f / sden[16 + l16];

    // ---------------- Phase C: pooled = segsum(mean_h(wn) * x) ----------------
    float accA = 0.f, accB = 0.f;   // lane owns output columns 2*lane, 2*lane+1
    for (int t = wave; t < ntiles; t += 8) {
        const int rowbase = start + t * 16;
        v8f c0, c1;
        compute_gates(rowbase, t + 8 < ntiles, c0, c1);
        float rs[8];
        #pragma unroll
        for (int i = 0; i < 8; ++i) {
            const bool v = rowbase + i + 8 * half < end;   // uniform per 16-lane half
            rs[i] = v ? (__expf(c0[i] - g0max) * rd0 + __expf(c1[i] - g1max) * rd1)
                      : 0.f;
        }
        // sum over the 16 lanes of each half -> per-row total over all 32 heads
        #pragma unroll
        for (int i = 0; i < 8; ++i) {
            rs[i] += __shfl_xor(rs[i], 1);
            rs[i] += __shfl_xor(rs[i], 2);
            rs[i] += __shfl_xor(rs[i], 4);
            rs[i] += __shfl_xor(rs[i], 8);
        }
        if (l16 == 0) {   // lanes 0 and 16 publish rows 0..7 / 8..15 of the tile
            #pragma unroll
            for (int i = 0; i < 8; ++i)
                swbar[wave][half * 8 + i] = rs[i] * (1.f / 32.f);
        }
        // DS ops are in-order within a wave: safe to read swbar back immediately.
        // Padded rows have swbar == 0 exactly, so clamp the address and let the
        // zero weight mask them -> 16 unconditional clause-able b64 loads.
        #pragma unroll
        for (int r = 0; r < 16; ++r) {
            int row = rowbase + r;
            if (row >= end) row = end - 1;
            const float w = swbar[wave][r];
            const v2f xv = ((const v2f*)(x + (size_t)row * DIM))[lane]; // b64
            accA += w * xv.x;
            accB += w * xv.y;
        }
    }
    atomicAdd(&spool[2 * lane],     accA);
    atomicAdd(&spool[2 * lane + 1], accB);
    __syncthreads();

    if (tid < DIM) out[seg * DIM + tid] = fmaxf(spool[tid], 0.f);
}

// ---------------------------------------------------------------------------
extern "C" void kernel_launch(void* const* d_in, const int* in_sizes, int n_in,
                              void* d_out, int out_size, void* d_ws, size_t ws_size,
                              hipStream_t stream) {
    const float* x     = (const float*)d_in[0];
    const int*   batch = (const int*)d_in[1];
    const float* Wg    = (const float*)d_in[2];
    const float* bg    = (const float*)d_in[3];
    float*       out   = (float*)d_out;
    const int    N     = in_sizes[1];          // batch has N entries

    int* seg_start = (int*)d_ws;               // 1025 ints of scratch

    seg_bounds_kernel<<<(NSEG + 1 + 255) / 256, 256, 0, stream>>>(batch, N, seg_start);
    gattp_main_kernel<<<NSEG, 256, 0, stream>>>(x, Wg, bg, seg_start, out);
}